// RCNN_10368051052849
// MI455X (gfx1250) — compile-verified
//
#include <hip/hip_runtime.h>
#include <math.h>

// RCNN (bidirectional recurrent context + maxpool + FC) for MI455X / gfx1250.
//
//  1) gather:      emb_t[t][b][:] = table[inputs[b][t]][:]
//  2) precompute:  S_dir[i] = e_prev_dir[i] @ Ws_dir[i] -- 1024 independent
//                  [32,256]x[256,256] fp32 GEMMs via V_WMMA_F32_16X16X4_F32.
//                  One wave owns a full 32x16 output slab (two 16x16 tiles,
//                  two accumulators) so each B fragment is loaded ONCE and
//                  reused from registers for both M-tiles.
//  3) scan:        serial recurrence c_i = tanh(c_{i-1} @ W_i + S[i]).
//                  One 512-thread block per direction: 16 waves == 16 column
//                  slabs, each wave computes the full 32-row state for its
//                  slab (2 accumulators, shared B regs -> minimal 256KB/step
//                  weight traffic on the serial chain). c double-buffered in
//                  LDS (row stride 260 for conflict-free strided A reads),
//                  S loads hoisted ahead of the GEMM, full-coverage per-lane
//                  prefetch of step i+1's weight/S slabs (8+1 instructions,
//                  32 cache lines each), running max fused into the loop.
//  4) embmax:      middle third of the pooled vector.
//  5) fc:          [32,768] @ [768,16]^T + bias -> d_out (512 floats).

typedef float v2f __attribute__((ext_vector_type(2)));
typedef float v8f __attribute__((ext_vector_type(8)));

#define TT 512
#define BB 32
#define EE 256
#define NLAB 16

__device__ __forceinline__ v8f wmma_f32_k4(v8f acc, v2f a, v2f b) {
    // D = A(16x4,f32) * B(4x16,f32) + C(16x16,f32)
    return __builtin_amdgcn_wmma_f32_16x16x4_f32(
        /*neg_a=*/false, a, /*neg_b=*/false, b,
        /*c_mod=*/(short)0, acc, /*reuse_a=*/false, /*reuse_b=*/false);
}

// ---------------------------------------------------------------- gather ----
__global__ void rcnn_gather(const int* __restrict__ inputs,
                            const float* __restrict__ table,
                            float* __restrict__ emb_t) {
    int tb = blockIdx.x;          // t*BB + b
    int t  = tb >> 5;
    int b  = tb & 31;
    int idx = inputs[b * TT + t]; // inputs is [B, T]
    int e = threadIdx.x;
    emb_t[((size_t)t * BB + b) * EE + e] = table[(size_t)idx * EE + e];
}

// ------------------------------------------------- batched S = eprev @ Ws ---
// grid: 2048 blocks x 256 thr (8 waves) = 16384 waves = 2 dirs * 512 steps
// * 16 column slabs. Each wave: 32x16 output slab, K in chunks of 4.
__global__ void rcnn_precompute_s(const float* __restrict__ emb_t,
                                  const float* __restrict__ e_prev0,
                                  const float* __restrict__ e_foll0,
                                  const float* __restrict__ Wsl,
                                  const float* __restrict__ Wsr,
                                  float* __restrict__ S_l,
                                  float* __restrict__ S_r) {
    unsigned lane = threadIdx.x & 31u;          // wave32
    unsigned wave = threadIdx.x >> 5;
    unsigned gw   = blockIdx.x * 8u + wave;
    unsigned nt   = gw & 15u;                   // column slab 0..15
    unsigned gemm = gw >> 4;
    unsigned i    = gemm & (TT - 1u);
    unsigned dir  = gemm >> 9;                  // 0 = left, 1 = right

    unsigned l16  = lane & 15u;
    unsigned half = lane >> 4;                  // lane-half -> K subgroup
    unsigned koff = half * 2u;
    unsigned n    = nt * 16u + l16;

    // e_prev sequence: left trails emb_t by one; right trails reversed emb_t.
    const float* aBase;
    if (dir == 0) aBase = (i == 0) ? e_prev0 : emb_t + (size_t)(i - 1) * BB * EE;
    else          aBase = (i == 0) ? e_foll0 : emb_t + (size_t)(TT - i) * BB * EE;
    const float* aRow0 = aBase + (size_t)l16 * EE;          // rows 0..15
    const float* aRow1 = aBase + (size_t)(16u + l16) * EE;  // rows 16..31
    const float* W     = ((dir == 0) ? Wsl : Wsr) + (size_t)i * EE * EE;

    v8f acc0 = {}, acc1 = {};
    #pragma unroll 4
    for (int k0 = 0; k0 < EE; k0 += 4) {
        v2f b;                                        // B loaded once,
        b.x = W[(size_t)(k0 + koff) * EE + n];        // reused for both tiles
        b.y = W[(size_t)(k0 + koff + 1) * EE + n];
        v2f a0 = *(const v2f*)(aRow0 + k0 + koff);
        v2f a1 = *(const v2f*)(aRow1 + k0 + koff);
        acc0 = wmma_f32_k4(acc0, a0, b);
        acc1 = wmma_f32_k4(acc1, a1, b);
    }

    float* S = ((dir == 0) ? S_l : S_r) + (size_t)i * BB * EE;
    #pragma unroll
    for (int v = 0; v < 8; ++v) {
        unsigned mr = half * 8u + v;                  // C/D VGPR v -> row
        S[(size_t)mr * EE + n] = acc0[v];
        S[(size_t)(16u + mr) * EE + n] = acc1[v];
    }
}

// ------------------------------------------------------- serial scan --------
// blockIdx.x = direction. 512 threads = 16 waves = 16 column slabs of the
// [32,256] state; each wave owns all 32 rows of its slab.
__global__ __launch_bounds__(512, 1)
void rcnn_scan(const float* __restrict__ Wl,
               const float* __restrict__ Wr,
               const float* __restrict__ S_l,
               const float* __restrict__ S_r,
               float* __restrict__ pooled) {       // [BB][3*EE]
    __shared__ float cbuf[2][BB][EE + 4];          // +4 pad: conflict-free A reads

    const int dir = blockIdx.x;
    const float* Wbase = (dir == 0) ? Wl : Wr;
    const float* Sbase = (dir == 0) ? S_l : S_r;
    const int poolcol0 = (dir == 0) ? 0 : 2 * EE;  // left | right third

    for (int idx = threadIdx.x; idx < BB * (EE + 4); idx += 512)
        (&cbuf[0][0][0])[idx] = 0.0f;              // c_0 = 0
    __syncthreads();

    unsigned lane = threadIdx.x & 31u;
    unsigned nt   = threadIdx.x >> 5;              // 0..15 column slab
    unsigned l16  = lane & 15u;
    unsigned half = lane >> 4;
    unsigned koff = half * 2u;
    unsigned n    = nt * 16u + l16;

    float rmax0[8], rmax1[8];
    #pragma unroll
    for (int v = 0; v < 8; ++v) { rmax0[v] = -3.402823466e+38f; rmax1[v] = rmax0[v]; }

    int cur = 0;
    for (int i = 0; i < TT; ++i) {
        const float* W  = Wbase + (size_t)i * EE * EE;
        const float* S  = Sbase + (size_t)i * BB * EE;

        // ---- one-step-ahead prefetch along the serial chain ----
        // Per-lane addresses: each instruction touches 32 distinct rows
        // (cache lines) of next step's weight slab -> 8 instructions cover
        // all 256 rows of this wave's 16-column slab; +1 for the S slab.
        {
            int ip1 = (i + 1 < TT) ? i + 1 : i;
            const float* Wn = Wbase + (size_t)ip1 * EE * EE;
            const float* Sn = Sbase + (size_t)ip1 * BB * EE;
            #pragma unroll
            for (int j = 0; j < 8; ++j)
                __builtin_prefetch(&Wn[(size_t)(j * 32u + lane) * EE + nt * 16u], 0, 1);
            __builtin_prefetch(&Sn[(size_t)lane * EE + nt * 16u], 0, 1);
        }

        // Hoist the S bias loads so they overlap the GEMM.
        float s0[8], s1[8];
        #pragma unroll
        for (int v = 0; v < 8; ++v) {
            unsigned mr = half * 8u + v;
            s0[v] = S[(size_t)mr * EE + n];
            s1[v] = S[(size_t)(16u + mr) * EE + n];
        }

        const float* aRow0 = &cbuf[cur][l16][0];
        const float* aRow1 = &cbuf[cur][16u + l16][0];
        v8f acc0 = {}, acc1 = {};
        #pragma unroll 4
        for (int k0 = 0; k0 < EE; k0 += 4) {
            v2f b;                                    // one B fetch per k-step
            b.x = W[(size_t)(k0 + koff) * EE + n];
            b.y = W[(size_t)(k0 + koff + 1) * EE + n];
            v2f a0 = *(const v2f*)(aRow0 + k0 + koff);   // LDS, 8B
            v2f a1 = *(const v2f*)(aRow1 + k0 + koff);
            acc0 = wmma_f32_k4(acc0, a0, b);
            acc1 = wmma_f32_k4(acc1, a1, b);
        }

        int nxt = cur ^ 1;
        #pragma unroll
        for (int v = 0; v < 8; ++v) {
            unsigned mr = half * 8u + v;
            float v0 = tanhf(acc0[v] + s0[v]);
            float v1 = tanhf(acc1[v] + s1[v]);
            cbuf[nxt][mr][n] = v0;
            cbuf[nxt][16u + mr][n] = v1;
            rmax0[v] = fmaxf(rmax0[v], v0);        // fused max-pool over t
            rmax1[v] = fmaxf(rmax1[v], v1);
        }
        __syncthreads();
        cur = nxt;
    }

    #pragma unroll
    for (int v = 0; v < 8; ++v) {
        unsigned mr = half * 8u + v;
        pooled[(size_t)mr * (3 * EE) + poolcol0 + n] = rmax0[v];
        pooled[(size_t)(16u + mr) * (3 * EE) + poolcol0 + n] = rmax1[v];
    }
}

// ----------------------------------------------- max over t of embeddings --
__global__ void rcnn_embmax(const float* __restrict__ emb_t,
                            float* __restrict__ pooled) {
    int g = blockIdx.x * blockDim.x + threadIdx.x;   // 0 .. BB*EE-1
    int b = g >> 8;
    int col = g & 255;
    float m = -3.402823466e+38f;
    for (int t = 0; t < TT; ++t)
        m = fmaxf(m, emb_t[((size_t)t * BB + b) * EE + col]);
    pooled[(size_t)b * (3 * EE) + EE + col] = m;
}

// ----------------------------------------------------------------- FC ------
__global__ void rcnn_fc(const float* __restrict__ pooled,
                        const float* __restrict__ fc_w,   // [16][768]
                        const float* __restrict__ fc_b,
                        float* __restrict__ out) {        // [32][16]
    int tid = threadIdx.x;        // 0..511
    int b = tid >> 4;
    int l = tid & 15;
    float s = fc_b[l];
    for (int j = 0; j < 3 * EE; ++j)
        s += pooled[(size_t)b * (3 * EE) + j] * fc_w[(size_t)l * (3 * EE) + j];
    out[b * NLAB + l] = s;
}

// ---------------------------------------------------------------- launch ----
extern "C" void kernel_launch(void* const* d_in, const int* in_sizes, int n_in,
                              void* d_out, int out_size, void* d_ws, size_t ws_size,
                              hipStream_t stream) {
    const int*   inputs    = (const int*)  d_in[0];   // [B,T]
    const float* emb_table = (const float*)d_in[1];   // [VOCAB,E]
    const float* Wl        = (const float*)d_in[2];   // [T,E,E]
    const float* Wsl       = (const float*)d_in[3];
    const float* Wr        = (const float*)d_in[4];
    const float* Wsr       = (const float*)d_in[5];
    const float* e_prev0   = (const float*)d_in[6];   // [B,E]
    const float* e_foll0   = (const float*)d_in[7];
    const float* fc_w      = (const float*)d_in[8];   // [16,768]
    const float* fc_b      = (const float*)d_in[9];   // [16]
    float* out = (float*)d_out;

    // Workspace layout (floats): needs ~50.4 MB.
    float* ws     = (float*)d_ws;
    float* emb_t  = ws;                             // [T][B][E] = 4,194,304
    float* S_l    = emb_t + (size_t)TT * BB * EE;   // 4,194,304
    float* S_r    = S_l   + (size_t)TT * BB * EE;   // 4,194,304
    float* pooled = S_r   + (size_t)TT * BB * EE;   // [B][3E]  = 24,576

    rcnn_gather      <<<TT * BB, EE, 0, stream>>>(inputs, emb_table, emb_t);
    rcnn_precompute_s<<<2048, 256, 0, stream>>>(emb_t, e_prev0, e_foll0,
                                                Wsl, Wsr, S_l, S_r);
    rcnn_scan        <<<2, 512, 0, stream>>>(Wl, Wr, S_l, S_r, pooled);
    rcnn_embmax      <<<(BB * EE) / 256, 256, 0, stream>>>(emb_t, pooled);
    rcnn_fc          <<<1, 512, 0, stream>>>(pooled, fc_w, fc_b, out);
}